// Net_48988396978415
// MI455X (gfx1250) — compile-verified
//
#include <hip/hip_runtime.h>
#include <hip/hip_bf16.h>

typedef float v2f __attribute__((ext_vector_type(2)));
typedef float v8f __attribute__((ext_vector_type(8)));

// ---------------------------------------------------------------- utilities
__global__ void fill_kernel(float* __restrict__ p, float v, int n) {
  int i = blockIdx.x * blockDim.x + threadIdx.x;
  if (i < n) p[i] = v;
}

__global__ void deg_scatter_kernel(const int* __restrict__ dst,
                                   const float* __restrict__ ew,
                                   float* __restrict__ deg, int E) {
  int e = blockIdx.x * blockDim.x + threadIdx.x;
  if (e < E) atomicAdd(&deg[dst[e]], ew[e]);
}

// in: p holds deg (>=1 always); out: p = rsqrt(deg), p2 = rsqrt(deg)^2
__global__ void dinv_kernel(float* __restrict__ p, float* __restrict__ p2, int n) {
  int i = blockIdx.x * blockDim.x + threadIdx.x;
  if (i < n) {
    float r = rsqrtf(p[i]);
    p[i] = r;
    p2[i] = r * r;
  }
}

__global__ void norm_kernel(const int* __restrict__ src, const int* __restrict__ dst,
                            const float* __restrict__ ew, const float* __restrict__ dinv,
                            float* __restrict__ norm, int E) {
  int e = blockIdx.x * blockDim.x + threadIdx.x;
  if (e < E) norm[e] = dinv[src[e]] * ew[e] * dinv[dst[e]];
}

// ------------------------------------------------------- WMMA fp32 GEMM
// Fused GCN-layer front half:
//   h   = relu?(X) @ W                         (written to H, consumed by scatter)
//   agg = h * dinv2[node] + bias[f]            (self-loop + bias, written to AGG)
// All matrices row-major. One wave per 16-row M-tile computing the full
// 16 x ND strip. M-tiles == gridDim.x*2 exactly (2 waves/block), so every
// wave runs with EXEC all-1s as WMMA requires.
//
// Fragment layouts per CDNA5 ISA 7.12.2 (32-bit, wave32):
//   A 16x4 : VGPR0 = K0 (lanes0-15) / K2 (lanes16-31); VGPR1 = K1 / K3
//   B 4x16 : VGPR0 = row K0 (lanes0-15) / row K2 (lanes16-31); VGPR1 = K1 / K3
//   C/D 16x16: VGPR v = row v (lanes0-15) / row v+8 (lanes16-31)
template<int KD, int ND, bool RELU>
__global__ void __launch_bounds__(64)
gemm_wmma_fused(const float* __restrict__ X, const float* __restrict__ W,
                const float* __restrict__ dinv2, const float* __restrict__ bias,
                float* __restrict__ H, float* __restrict__ AGG) {
  const int lane = threadIdx.x & 31;
  const int wave = threadIdx.x >> 5;
  const int mt   = blockIdx.x * 2 + wave;   // M-tile index (uniform per wave)
  const int row  = lane & 15;
  const int kh   = lane >> 4;               // 0 or 1 (K half)
  constexpr int NT = ND / 16;

  v8f acc[NT] = {};
  const float* __restrict__ Xr = X + (size_t)(mt * 16 + row) * KD;

  for (int k = 0; k < KD; k += 4) {
    v2f a;
    if (RELU) {
      a.x = fmaxf(Xr[k + 2 * kh + 0], 0.0f);
      a.y = fmaxf(Xr[k + 2 * kh + 1], 0.0f);
    } else {
      a.x = Xr[k + 2 * kh + 0];
      a.y = Xr[k + 2 * kh + 1];
    }
#pragma unroll
    for (int nt = 0; nt < NT; ++nt) {
      const int col = nt * 16 + row;
      v2f b;
      b.x = W[(size_t)(k + 2 * kh + 0) * ND + col];
      b.y = W[(size_t)(k + 2 * kh + 1) * ND + col];
      acc[nt] = __builtin_amdgcn_wmma_f32_16x16x4_f32(
          /*neg_a=*/false, a, /*neg_b=*/false, b,
          /*c_mod=*/(short)0, acc[nt], /*reuse_a=*/false, /*reuse_b=*/false);
    }
  }

  // epilogue: write h and the self-loop+bias initialized agg
  float d2[8];
#pragma unroll
  for (int v = 0; v < 8; ++v) d2[v] = dinv2[mt * 16 + v + 8 * kh];

#pragma unroll
  for (int nt = 0; nt < NT; ++nt) {
    const int col = nt * 16 + row;
    const float bv = bias[col];
#pragma unroll
    for (int v = 0; v < 8; ++v) {
      const size_t idx = (size_t)(mt * 16 + v + 8 * kh) * ND + col;
      const float hv = acc[nt][v];
      H[idx]   = hv;
      AGG[idx] = hv * d2[v] + bv;
    }
  }
}

// agg[dst] += h[src] * norm   — one wave per edge, lanes stride features
template<int F>
__global__ void edge_scatter_kernel(const float* __restrict__ h,
                                    const float* __restrict__ norm,
                                    const int* __restrict__ src,
                                    const int* __restrict__ dst,
                                    float* __restrict__ agg, int E) {
  int e = blockIdx.x * (blockDim.x >> 5) + (threadIdx.x >> 5);
  if (e >= E) return;
  int lane = threadIdx.x & 31;
  int s = src[e], d = dst[e];
  float w = norm[e];
#pragma unroll
  for (int f = lane; f < F; f += 32)
    atomicAdd(&agg[(size_t)d * F + f], h[(size_t)s * F + f] * w);
}

// max over nodes of relu(X[n][f]) for F=32. Pool init is 0 and we fold with
// fmaxf, so pre-relu input gives the identical result (max(0, max(agg))).
// Values are >= 0 bitwise-monotone, so int atomicMax is exact.
__global__ void pool_max32_kernel(const float* __restrict__ X,
                                  float* __restrict__ pool, int n) {
  __shared__ float lds[256];
  int t = threadIdx.x;
  int f = t & 31;
  int g = t >> 5;   // 0..7
  float m = 0.0f;
  for (int i = blockIdx.x * 8 + g; i < n; i += gridDim.x * 8)
    m = fmaxf(m, X[(size_t)i * 32 + f]);
  lds[t] = m;
  __syncthreads();
  if (t < 32) {
    float mm = lds[t];
#pragma unroll
    for (int k = 1; k < 8; ++k) mm = fmaxf(mm, lds[k * 32 + t]);
    atomicMax((int*)(pool + t), __float_as_int(mm));
  }
}

// ------------------------------------------------------------- MLP head
__global__ void __launch_bounds__(128)
mlp_kernel(const float* __restrict__ pool0, const float* __restrict__ pool1,
           const float* __restrict__ C2ER,
           const float* __restrict__ L1W, const float* __restrict__ L1b,
           const float* __restrict__ L2W, const float* __restrict__ L2b,
           const float* __restrict__ L3W, const float* __restrict__ L3b,
           float* __restrict__ out) {
  __shared__ float code[68];
  __shared__ float z1[128];
  __shared__ float z2[128];
  __shared__ float logits[10];
  int t = threadIdx.x;
  if (t < 32)             code[t] = pool0[t];
  else if (t < 64)        code[t] = pool1[t - 32];
  else if (t < 68)        code[t] = C2ER[t - 64];
  __syncthreads();

  float s = L1b[t];
  for (int k = 0; k < 68; ++k) s += code[k] * L1W[k * 128 + t];
  z1[t] = fmaxf(s, 0.0f);
  __syncthreads();

  s = L2b[t];
  for (int k = 0; k < 128; ++k) s += z1[k] * L2W[k * 128 + t];
  z2[t] = fmaxf(s, 0.0f);
  __syncthreads();

  if (t < 10) {
    float l = L3b[t];
    for (int k = 0; k < 128; ++k) l += z2[k] * L3W[k * 10 + t];
    logits[t] = l;
  }
  __syncthreads();

  if (t == 0) {
    float mx = logits[0];
    for (int i = 1; i < 10; ++i) mx = fmaxf(mx, logits[i]);
    float ex[10], sum = 0.0f;
    for (int i = 0; i < 10; ++i) { ex[i] = expf(logits[i] - mx); sum += ex[i]; }
    for (int i = 0; i < 10; ++i) out[i] = ex[i] / sum;
  }
  if (t < 68) out[10 + t] = code[t];   // code follows the 10 softmax outputs
}

// ---------------------------------------------------------------- launch
extern "C" void kernel_launch(void* const* d_in, const int* in_sizes, int n_in,
                              void* d_out, int out_size, void* d_ws, size_t ws_size,
                              hipStream_t stream) {
  const float* x    = (const float*)d_in[0];
  const int*   ei   = (const int*)d_in[1];
  const float* ew   = (const float*)d_in[2];
  const float* C2ER = (const float*)d_in[4];
  const float* W1 = (const float*)d_in[5];  const float* b1 = (const float*)d_in[6];
  const float* W2 = (const float*)d_in[7];  const float* b2 = (const float*)d_in[8];
  const float* W3 = (const float*)d_in[9];  const float* b3 = (const float*)d_in[10];
  const float* W4 = (const float*)d_in[11]; const float* b4 = (const float*)d_in[12];
  const float* L1W = (const float*)d_in[13]; const float* L1b = (const float*)d_in[14];
  const float* L2W = (const float*)d_in[15]; const float* L2b = (const float*)d_in[16];
  const float* L3W = (const float*)d_in[17]; const float* L3b = (const float*)d_in[18];

  const int N = in_sizes[0] / 8;   // 100000
  const int E = in_sizes[2];       // 600000
  const int* src = ei;
  const int* dst = ei + E;

  // workspace carve-out (256B aligned)
  char* ws = (char*)d_ws;
  size_t off = 0;
  auto carve = [&](size_t bytes) -> float* {
    float* p = (float*)(ws + off);
    off = (off + bytes + 255) & ~(size_t)255;
    return p;
  };
  float* dinv  = carve((size_t)N * 4);
  float* dinv2 = carve((size_t)N * 4);
  float* normw = carve((size_t)E * 4);
  float* pool  = carve(64 * 4);            // pool0 = pool[0..31], pool1 = pool[32..63]
  float* buf0  = carve((size_t)N * 128 * 4);
  float* buf1  = carve((size_t)N * 128 * 4);
  float* buf2  = carve((size_t)N * 128 * 4);
  float* pool0 = pool;
  float* pool1 = pool + 32;

  const int TB = 256;
  auto blk = [&](long long n) { return (int)((n + TB - 1) / TB); };
  const int mtiles = N / 16;               // 6250
  const int gemm_blocks = mtiles / 2;      // 2 waves per 64-thread block
  const int escat = (E + 7) / 8;           // 8 waves (edges) per 256-thread block

  // degree / normalization (shared by all 4 convs)
  fill_kernel<<<blk(N), TB, 0, stream>>>(dinv, 1.0f, N);                 // self-loop
  fill_kernel<<<1, 64, 0, stream>>>(pool, 0.0f, 64);
  deg_scatter_kernel<<<blk(E), TB, 0, stream>>>(dst, ew, dinv, E);
  dinv_kernel<<<blk(N), TB, 0, stream>>>(dinv, dinv2, N);
  norm_kernel<<<blk(E), TB, 0, stream>>>(src, dst, ew, dinv, normw, E);

  // ---- layer 1: 8 -> 32 (input is raw x, no relu) ----
  gemm_wmma_fused<8, 32, false><<<gemm_blocks, 64, 0, stream>>>(x, W1, dinv2, b1, buf1, buf2);
  edge_scatter_kernel<32><<<escat, TB, 0, stream>>>(buf1, normw, src, dst, buf2, E);
  pool_max32_kernel<<<256, 256, 0, stream>>>(buf2, pool0, N);   // == max over relu

  // ---- layer 2: 32 -> 128 (relu applied on A-load) ----
  gemm_wmma_fused<32, 128, true><<<gemm_blocks, 64, 0, stream>>>(buf2, W2, dinv2, b2, buf0, buf1);
  edge_scatter_kernel<128><<<escat, TB, 0, stream>>>(buf0, normw, src, dst, buf1, E);

  // ---- layer 3: 128 -> 128 ----
  gemm_wmma_fused<128, 128, true><<<gemm_blocks, 64, 0, stream>>>(buf1, W3, dinv2, b3, buf2, buf0);
  edge_scatter_kernel<128><<<escat, TB, 0, stream>>>(buf2, normw, src, dst, buf0, E);

  // ---- layer 4: 128 -> 32 ----
  gemm_wmma_fused<128, 32, true><<<gemm_blocks, 64, 0, stream>>>(buf0, W4, dinv2, b4, buf1, buf2);
  edge_scatter_kernel<32><<<escat, TB, 0, stream>>>(buf1, normw, src, dst, buf2, E);
  pool_max32_kernel<<<256, 256, 0, stream>>>(buf2, pool1, N);   // == max over relu

  // ---- head ----
  mlp_kernel<<<1, 128, 0, stream>>>(pool0, pool1, C2ER,
                                    L1W, L1b, L2W, L2b, L3W, L3b,
                                    (float*)d_out);
}